// HebbianConv2d_50311246905730
// MI455X (gfx1250) — compile-verified
//
#include <hip/hip_runtime.h>
#include <stdint.h>

// ---------------- problem constants ----------------
#define B_     16
#define CIN    256
#define COUT   384
#define H_     56
#define W_     56
#define HOUT   54
#define HW_IN  (H_ * W_)        // 3136
#define CHW_IN (CIN * HW_IN)    // 802816
#define HWO    (HOUT * HOUT)    // 2916
#define NTOT   (B_ * HWO)       // 46656 (= 16 * 2916, tiles divide exactly)
#define PADC   258              // 256 ci + 2-float pad: lanes lo=0..15 -> distinct LDS banks

typedef __attribute__((ext_vector_type(2))) float v2f;
typedef __attribute__((ext_vector_type(8))) float v8f;

// ---------------------------------------------------------------------------
// Kernel 1: implicit-GEMM conv2d via V_WMMA_F32_16X16X4_F32 (fp32-exact),
// with per-tap activation tiles staged to LDS through the CDNA5 async path
// (GLOBAL_LOAD_ASYNC_TO_LDS_B32, tracked by ASYNCcnt), double-buffered.
//
// Block = 256 threads = 8 waves: 128 couts x 16 flattened output positions.
// K-loop: tap-major (9 taps) x ci (256, step 4) = 576 WMMAs per 16x16 tile.
//
// Operand striping per CDNA5 ISA 7.12.2:
//   A 16x4 f32 : lane&15 = M; VGPR0 -> K = 2*hi, VGPR1 -> K = 2*hi+1
//   B 4x16 f32 : lane&15 = N; VGPR0 -> K = 2*hi, VGPR1 -> K = 2*hi+1
//   C/D 16x16  : VGPR v, lanes 0-15 -> M=v, lanes 16-31 -> M=v+8, N = lane&15
// ---------------------------------------------------------------------------
__global__ __launch_bounds__(256) void conv_wmma_f32(
    const float* __restrict__ x,      // (B, CIN, 56, 56)
    const float* __restrict__ w,      // (COUT, CIN, 3, 3)
    float* __restrict__ y)            // (B, COUT, 54, 54)
{
    __shared__ __align__(16) float xs[2][16 * PADC];   // double-buffered tap tile

    const int tid  = threadIdx.x;                      // 0..255 == ci for staging
    const int lane = tid & 31;
    const int wave = tid >> 5;
    const int hi   = lane >> 4;                        // K-half select
    const int lo   = lane & 15;                        // M for A, N for B

    const int cout0 = blockIdx.y * 128 + wave * 16;

    // flattened output position owned by this lane's B/C columns
    const int n   = blockIdx.x * 16 + lo;
    const int b   = n / HWO;
    const int rem = n - b * HWO;
    const int oh  = rem / HOUT;
    const int ow  = rem - oh * HOUT;
    const int posOff = b * CHW_IN + oh * W_ + ow;      // lanes 0-15 hold the 16 tile bases

    const int    m    = cout0 + lo;
    const float* wrow = w + (long)m * (CIN * 9);

    // byte offset of xs in LDS (single static __shared__ object -> offset 0 base)
    const uint32_t ldsBase = (uint32_t)(uintptr_t)&xs[0][0];

    // ---- async-stage one tap's tile: xs[buf][pos][ci] = x[b, ci, oh+kh, ow+kw]
    auto stage = [&](int tap, int buf) {
        const int kh = tap / 3;
        const int kw = tap - kh * 3;
        const int tapoff = kh * W_ + kw;
        const uint32_t ldsBuf = ldsBase + (uint32_t)(buf * 16 * PADC) * 4u;
        #pragma unroll
        for (int p = 0; p < 16; ++p) {
            const int pbase = __shfl(posOff, p, 32);   // uniform: tile position p's base
            const uint64_t ga =
                (uint64_t)(uintptr_t)(x + (long)pbase + tapoff + (long)tid * HW_IN);
            const uint32_t la = ldsBuf + (uint32_t)(p * PADC + tid) * 4u;
            // per-lane gather straight into LDS; tracked by ASYNCcnt
            asm volatile("global_load_async_to_lds_b32 %0, %1, off"
                         :: "v"(la), "v"(ga) : "memory");
        }
    };

    stage(0, 0);                                        // prologue fill

    v8f acc = {};                                       // 16x16 f32 accumulator
    for (int tap = 0; tap < 9; ++tap) {
        // this wave's async loads for buf[tap&1] done, then block-wide visibility
        asm volatile("s_wait_asynccnt 0x0" ::: "memory");
        __syncthreads();

        if (tap + 1 < 9) stage(tap + 1, (tap + 1) & 1); // prefetch next tap

        const float* xrow = &xs[tap & 1][lo * PADC];
        #pragma unroll 8
        for (int q = 0; q < CIN; q += 4) {
            const int ci = q + 2 * hi;                  // K = 2*hi, 2*hi+1
            v2f a;
            a[0] = wrow[ci * 9 + tap];                  // L2-resident weight gather
            a[1] = wrow[(ci + 1) * 9 + tap];
            // one bank-conflict-free ds_load_b64 feeds the B operand
            const v2f bmat =
                *(const v2f*)__builtin_assume_aligned(xrow + ci, 8);
            acc = __builtin_amdgcn_wmma_f32_16x16x4_f32(
                      false, a, false, bmat, (short)0, acc, false, false);
        }
        __syncthreads();   // all waves done reading buf[tap&1] before it is re-staged
    }

    // store into NCHW output: channel c = cout0 + v + 8*hi, column = this lane's n
    const long obase = (long)b * COUT * HWO + (long)oh * HOUT + ow;
    #pragma unroll
    for (int v = 0; v < 8; ++v) {
        const int c = cout0 + v + 8 * hi;
        y[obase + (long)c * HWO] = acc[v];
    }
}

// ---------------------------------------------------------------------------
// Kernel 2: winner-take-all + Gaussian lateral feedback + gating, in place.
// One wave32 per row (384 channels, 12 per lane): shuffle max-reduce, winner
// indices appended to an LDS list (typically 1), then
// lfb[c] = clip(sum_j exp(-(c-j)^2/(2*191^2)), -1, 1); y[c] *= lfb[c].
// ---------------------------------------------------------------------------
__global__ __launch_bounds__(256) void wta_lfb_gate(float* __restrict__ y)
{
    __shared__ int s_cnt[8];
    __shared__ int s_win[8][COUT];

    const int lane = threadIdx.x & 31;
    const int wave = threadIdx.x >> 5;

    const int n   = blockIdx.x * 8 + wave;
    const int b   = n / HWO;
    const int rem = n - b * HWO;
    const long base = (long)b * COUT * HWO + rem;       // + c*HWO -> channel c

    float vals[12];
    float mx = -INFINITY;
    #pragma unroll
    for (int i = 0; i < 12; ++i) {
        const int c = lane + 32 * i;
        vals[i] = y[base + (long)c * HWO];
        mx = fmaxf(mx, vals[i]);
    }
    #pragma unroll
    for (int off = 16; off >= 1; off >>= 1)             // wave32 max reduction
        mx = fmaxf(mx, __shfl_xor(mx, off, 32));

    if (lane == 0) s_cnt[wave] = 0;
    __syncthreads();

    #pragma unroll
    for (int i = 0; i < 12; ++i) {
        if (vals[i] >= mx) {                            // same compare as reference
            const int idx = atomicAdd(&s_cnt[wave], 1);
            s_win[wave][idx] = lane + 32 * i;
        }
    }
    __syncthreads();

    const int   cnt    = s_cnt[wave];
    const float inv2s2 = 1.0f / (2.0f * 191.0f * 191.0f);

    #pragma unroll
    for (int i = 0; i < 12; ++i) {
        const int c = lane + 32 * i;
        float lfb = 0.0f;
        for (int j = 0; j < cnt; ++j) {
            const float d = (float)(c - s_win[wave][j]);
            lfb += __expf(-d * d * inv2s2);
        }
        lfb = fminf(fmaxf(lfb, -1.0f), 1.0f);
        y[base + (long)c * HWO] = lfb * vals[i];
    }
}

// ---------------------------------------------------------------------------
extern "C" void kernel_launch(void* const* d_in, const int* in_sizes, int n_in,
                              void* d_out, int out_size, void* d_ws, size_t ws_size,
                              hipStream_t stream)
{
    const float* x   = (const float*)d_in[0];   // (16,256,56,56) fp32
    const float* w   = (const float*)d_in[1];   // (384,256,3,3)  fp32
    float*       out = (float*)d_out;           // (16,384,54,54) fp32

    dim3 g1(NTOT / 16, 3), b1(256);
    conv_wmma_f32<<<g1, b1, 0, stream>>>(x, w, out);

    dim3 g2(NTOT / 8), b2(256);
    wta_lfb_gate<<<g2, b2, 0, stream>>>(out);
}